// Node_Decoder_34832184770746
// MI455X (gfx1250) — compile-verified
//
#include <hip/hip_runtime.h>
#include <hip/hip_bf16.h>

// ---------------------------------------------------------------------------
// Fully-fused per-group GNN decoder for MI455X (gfx1250, wave32, WMMA).
// One workgroup (128 threads = 4 wave32) per group of 32 nodes.
// All intermediates (latent tile, x, 1024x8 edge features, weights) in LDS.
// All GEMMs via V_WMMA_F32_16X16X4_F32 (fp32, matches reference precision).
// Round 2: loop-invariant B fragments / biases hoisted to registers in the
// hot edge loops; edge A-fragment gathers split into branch-free phases.
// ---------------------------------------------------------------------------

typedef __attribute__((ext_vector_type(2))) float v2f;
typedef __attribute__((ext_vector_type(8))) float v8f;

#define SLOPE 0.01f

__device__ __forceinline__ float lrelu(float v) { return v > 0.f ? v : SLOPE * v; }

#if __has_builtin(__builtin_amdgcn_wmma_f32_16x16x4_f32)
#define HAVE_WMMA_F32 1
#endif

__device__ __forceinline__ v8f wmma4(v2f a, v2f b, v8f c) {
#ifdef HAVE_WMMA_F32
  // 8 args: (neg_a, A, neg_b, B, c_mod, C, reuse_a, reuse_b)
  return __builtin_amdgcn_wmma_f32_16x16x4_f32(false, a, false, b, (short)0, c,
                                               false, false);
#else
  // Fallback so a missing builtin still compiles (will show wmma=0).
  c[0] += a.x * b.x + a.y * b.y;
  return c;
#endif
}

// LDS layout (floats)
#define O_LAT 0       // 32x64 latent tile            (2048)
#define O_X 2048      // 32x16 node features          (512)
#define O_M 2560      // 32x8  aggregated messages    (256)
#define O_E 2816      // 1024x8 edge features         (8192)
#define O_SCR 11008   // 32x32 staging scratch        (1024)
#define O_LTW1 12032  // 64x32                        (2048)
#define O_LTW2 14080  // 32x16                        (512)
#define O_WE0 14592   // 40x16 padded                 (640)
#define O_WE1 15232   // 40x16 padded                 (640)
#define O_BE 15872    // be0(8) be1(8)                (16)
#define O_WN0 15888   // 24x16                        (384)
#define O_WN1 16272   // 24x16                        (384)
#define O_BN 16656    // bn0(16) bn1(16)              (32)
#define O_ETW1 16688  // 8x16                         (128)
#define O_ETB1 16816  // 16                           (16)
#define O_ETW2 16832  // 16x16 padded (cols>=5 -> 0)  (256)
#define O_ETB2 17088  // 16 padded                    (16)
#define O_NFW1 17104  // 16x32                        (512)
#define O_NFB1 17616  // 32                           (32)
#define O_NFW2 17648  // 32x16 padded (cols>=11 -> 0) (512)
#define O_NFB2 18160  // 16 padded                    (16)
#define SMEM_FLOATS 18176

__global__ void __launch_bounds__(128) gnn_fused(
    const float* __restrict__ latent, const float* __restrict__ ltw1,
    const float* __restrict__ ltw2, const float* __restrict__ We0,
    const float* __restrict__ be0, const float* __restrict__ We1,
    const float* __restrict__ be1, const float* __restrict__ Wn0,
    const float* __restrict__ bn0, const float* __restrict__ Wn1,
    const float* __restrict__ bn1, const float* __restrict__ etw1,
    const float* __restrict__ etb1, const float* __restrict__ etw2,
    const float* __restrict__ etb2, const float* __restrict__ nfw1,
    const float* __restrict__ nfb1, const float* __restrict__ nfw2,
    const float* __restrict__ nfb2, float* __restrict__ out_node,
    float* __restrict__ out_edge) {
  extern __shared__ float sm[];
  const int g = blockIdx.x;
  const int tid = threadIdx.x;
  const int w = tid >> 5;      // wave id 0..3
  const int lane = tid & 31;   // lane in wave32
  const int half = lane >> 4;  // K-half for A/B fragments
  const int lr = lane & 15;    // row (A) / col (B,C) index

  // ---- cooperative loads: latent tile + all weights (zero-padded) ----
  for (int i = tid; i < 2048; i += 128)
    sm[O_LAT + i] = latent[(size_t)g * 2048 + i];
  for (int i = tid; i < 2048; i += 128) sm[O_LTW1 + i] = ltw1[i];
  for (int i = tid; i < 512; i += 128) sm[O_LTW2 + i] = ltw2[i];
  for (int i = tid; i < 640; i += 128) {
    const int r = i >> 4, c = i & 15;
    sm[O_WE0 + i] = (c < 8 && r < 37) ? We0[r * 8 + c] : 0.f;
    sm[O_WE1 + i] = (c < 8) ? We1[r * 8 + c] : 0.f;
  }
  if (tid < 8) {
    sm[O_BE + tid] = be0[tid];
    sm[O_BE + 8 + tid] = be1[tid];
  }
  for (int i = tid; i < 384; i += 128) {
    sm[O_WN0 + i] = Wn0[i];
    sm[O_WN1 + i] = Wn1[i];
  }
  if (tid < 16) {
    sm[O_BN + tid] = bn0[tid];
    sm[O_BN + 16 + tid] = bn1[tid];
  }
  for (int i = tid; i < 128; i += 128) sm[O_ETW1 + i] = etw1[i];
  if (tid < 16) sm[O_ETB1 + tid] = etb1[tid];
  for (int i = tid; i < 256; i += 128) {
    const int r = i >> 4, c = i & 15;
    sm[O_ETW2 + i] = (c < 5) ? etw2[r * 5 + c] : 0.f;
  }
  if (tid < 16) sm[O_ETB2 + tid] = (tid < 5) ? etb2[tid] : 0.f;
  for (int i = tid; i < 512; i += 128) sm[O_NFW1 + i] = nfw1[i];
  if (tid < 32) sm[O_NFB1 + tid] = nfb1[tid];
  for (int i = tid; i < 512; i += 128) {
    const int r = i >> 4, c = i & 15;
    sm[O_NFW2 + i] = (c < 11) ? nfw2[r * 11 + c] : 0.f;
  }
  if (tid < 16) sm[O_NFB2 + tid] = (tid < 11) ? nfb2[tid] : 0.f;
  __syncthreads();

  // ---- stage 1: x = lrelu(lat @ ltw1) @ ltw2  ([32x64]@[64x32]@[32x16]) ----
  {
    const int mt = w >> 1, nt = w & 1;  // 2 M-tiles x 2 N-tiles over 4 waves
    v8f acc = {};
#pragma unroll
    for (int kk = 0; kk < 16; ++kk) {
      const int k0 = kk * 4 + half * 2;
      v2f a, b;
      a.x = sm[O_LAT + (mt * 16 + lr) * 64 + k0];
      a.y = sm[O_LAT + (mt * 16 + lr) * 64 + k0 + 1];
      b.x = sm[O_LTW1 + k0 * 32 + nt * 16 + lr];
      b.y = sm[O_LTW1 + (k0 + 1) * 32 + nt * 16 + lr];
      acc = wmma4(a, b, acc);
    }
#pragma unroll
    for (int j = 0; j < 8; ++j)
      sm[O_SCR + (mt * 16 + j + 8 * half) * 32 + nt * 16 + lr] = lrelu(acc[j]);
  }
  __syncthreads();
  if (w < 2) {
    v8f acc = {};
#pragma unroll
    for (int kk = 0; kk < 8; ++kk) {
      const int k0 = kk * 4 + half * 2;
      v2f a, b;
      a.x = sm[O_SCR + (w * 16 + lr) * 32 + k0];
      a.y = sm[O_SCR + (w * 16 + lr) * 32 + k0 + 1];
      b.x = sm[O_LTW2 + k0 * 16 + lr];
      b.y = sm[O_LTW2 + (k0 + 1) * 16 + lr];
      acc = wmma4(a, b, acc);
    }
#pragma unroll
    for (int j = 0; j < 8; ++j)
      sm[O_X + (w * 16 + j + 8 * half) * 16 + lr] = acc[j];
  }
  __syncthreads();

  // ---- stage 2: three edge-conv applications (layers 0,1 relu; 1 no-relu) --
  const float* sx = sm + O_X;
  const float* se = sm + O_E;
  for (int it = 0; it < 3; ++it) {
    const int l = (it == 0) ? 0 : 1;
    const bool dorelu = (it < 2);
    const int oWe = l ? O_WE1 : O_WE0;
    const int oBe = O_BE + (l ? 8 : 0);
    const int oWn = l ? O_WN1 : O_WN0;
    const int oBn = O_BN + (l ? 16 : 0);

    // Hoist loop-invariant B fragments (We, 10 K-steps) + bias to registers.
    v2f bw[10];
#pragma unroll
    for (int kk = 0; kk < 10; ++kk) {
      const int k0 = kk * 4 + half * 2;
      bw[kk].x = sm[oWe + k0 * 16 + lr];
      bw[kk].y = sm[oWe + (k0 + 1) * 16 + lr];
    }
    const float bias_e = sm[oBe + (lr & 7)];

    // edge features: 64 tiles of 16 edges, [16x40(pad)] @ [40x16(pad)]
    for (int i = 0; i < 16; ++i) {
      const int t = i * 4 + w;
      const int eid0 = t * 16;
      const int eid = eid0 + lr;  // A row this lane supplies
      const int d = eid >> 5, s = eid & 31;
      const float* xs = sx + s * 16;
      const float* xd = sx + d * 16;
      v8f acc = {};
#pragma unroll
      for (int kk = 0; kk < 4; ++kk) {  // K 0..15: x[src]
        const int k0 = kk * 4 + half * 2;
        v2f a;
        a.x = xs[k0];
        a.y = xs[k0 + 1];
        acc = wmma4(a, bw[kk], acc);
      }
#pragma unroll
      for (int kk = 4; kk < 8; ++kk) {  // K 16..31: x[dst]
        const int k0 = (kk - 4) * 4 + half * 2;
        v2f a;
        a.x = xd[k0];
        a.y = xd[k0 + 1];
        acc = wmma4(a, bw[kk], acc);
      }
      if (it == 0) {  // K 32..39: attr = self-loop pattern [0,1,1,1,1]
#pragma unroll
        for (int kk = 8; kk < 10; ++kk) {
          const int ka = (kk - 8) * 4 + half * 2;
          v2f a;
          a.x = (s == d && ka >= 1 && ka < 5) ? 1.f : 0.f;
          a.y = (s == d && ka + 1 >= 1 && ka + 1 < 5) ? 1.f : 0.f;
          acc = wmma4(a, bw[kk], acc);
        }
      } else {  // K 32..39: attr = previous-layer edge features (8 wide)
#pragma unroll
        for (int kk = 8; kk < 10; ++kk) {
          const int ka = (kk - 8) * 4 + half * 2;
          v2f a;
          a.x = se[eid * 8 + ka];
          a.y = se[eid * 8 + ka + 1];
          acc = wmma4(a, bw[kk], acc);
        }
      }
      if (lr < 8) {
#pragma unroll
        for (int j = 0; j < 8; ++j)
          sm[O_E + (eid0 + j + 8 * half) * 8 + lr] = lrelu(acc[j] + bias_e);
      }
    }
    __syncthreads();

    // segment sum: m[d][c] = sum_s e[(d,s)][c]  (every dst sees all 32 srcs)
    for (int p = tid; p < 256; p += 128) {
      const int d2 = p >> 3, c = p & 7;
      float sum = 0.f;
#pragma unroll 8
      for (int s2 = 0; s2 < 32; ++s2) sum += sm[O_E + (d2 * 32 + s2) * 8 + c];
      sm[O_M + d2 * 8 + c] = sum;
    }
    __syncthreads();

    // node update: x = concat(x, m) @ Wn + bn   ([32x24] @ [24x16])
    if (w < 2) {
      v8f acc = {};
#pragma unroll
      for (int kk = 0; kk < 6; ++kk) {
        const int k0 = kk * 4 + half * 2;
        const int r = w * 16 + lr;
        v2f a, b;
        a.x = (k0 < 16) ? sm[O_X + r * 16 + k0] : sm[O_M + r * 8 + (k0 - 16)];
        a.y = (k0 + 1 < 16) ? sm[O_X + r * 16 + k0 + 1]
                            : sm[O_M + r * 8 + (k0 + 1 - 16)];
        b.x = sm[oWn + k0 * 16 + lr];
        b.y = sm[oWn + (k0 + 1) * 16 + lr];
        acc = wmma4(a, b, acc);
      }
#pragma unroll
      for (int j = 0; j < 8; ++j) {
        float v = acc[j] + sm[oBn + lr];
        if (dorelu) v = fmaxf(v, 0.f);
        sm[O_X + (w * 16 + j + 8 * half) * 16 + lr] = v;
      }
    }
    __syncthreads();
  }

  // ---- stage 3: edge head on final e: lrelu(e@etw1+b1)@etw2+b2 ----
  {
    // Hoist invariant B fragments and biases.
    v2f bw1[2], bw2[4];
#pragma unroll
    for (int kk = 0; kk < 2; ++kk) {
      const int k0 = kk * 4 + half * 2;
      bw1[kk].x = sm[O_ETW1 + k0 * 16 + lr];
      bw1[kk].y = sm[O_ETW1 + (k0 + 1) * 16 + lr];
    }
#pragma unroll
    for (int kk = 0; kk < 4; ++kk) {
      const int k0 = kk * 4 + half * 2;
      bw2[kk].x = sm[O_ETW2 + k0 * 16 + lr];
      bw2[kk].y = sm[O_ETW2 + (k0 + 1) * 16 + lr];
    }
    const float eb1 = sm[O_ETB1 + lr];
    const float eb2 = sm[O_ETB2 + lr];

    for (int i = 0; i < 16; ++i) {
      const int t = i * 4 + w;
      const int eid0 = t * 16;
      v8f acc = {};
#pragma unroll
      for (int kk = 0; kk < 2; ++kk) {  // K = 8
        const int k0 = kk * 4 + half * 2;
        v2f a;
        a.x = sm[O_E + (eid0 + lr) * 8 + k0];
        a.y = sm[O_E + (eid0 + lr) * 8 + k0 + 1];
        acc = wmma4(a, bw1[kk], acc);
      }
      float* scr = sm + O_SCR + w * 256;  // per-wave 16x16 staging
#pragma unroll
      for (int j = 0; j < 8; ++j)
        scr[(j + 8 * half) * 16 + lr] = lrelu(acc[j] + eb1);
      v8f acc2 = {};
#pragma unroll
      for (int kk = 0; kk < 4; ++kk) {  // K = 16
        const int k0 = kk * 4 + half * 2;
        v2f a;
        a.x = scr[lr * 16 + k0];
        a.y = scr[lr * 16 + k0 + 1];
        acc2 = wmma4(a, bw2[kk], acc2);
      }
      if (lr < 5) {
#pragma unroll
        for (int j = 0; j < 8; ++j) {
          const int eid = eid0 + j + 8 * half;
          const int d = eid >> 5, s2 = eid & 31;
          if (s2 != d) {  // self-loop edges are dropped (attr[:-N])
            const int k = (d - s2) & 31;  // 1..31
            out_edge[((size_t)g * 992 + (size_t)(s2 * 31 + k - 1)) * 5 + lr] =
                acc2[j] + eb2;
          }
        }
      }
    }
  }
  __syncthreads();

  // ---- stage 4: node head: lrelu(x@nfw1+b1)@nfw2+b2 ----
  if (w < 2) {
    float* scr = sm + O_SCR + w * 512;  // per-wave 16x32 staging
#pragma unroll
    for (int nt = 0; nt < 2; ++nt) {
      v8f acc = {};
#pragma unroll
      for (int kk = 0; kk < 4; ++kk) {  // K = 16
        const int k0 = kk * 4 + half * 2;
        v2f a, b;
        a.x = sm[O_X + (w * 16 + lr) * 16 + k0];
        a.y = sm[O_X + (w * 16 + lr) * 16 + k0 + 1];
        b.x = sm[O_NFW1 + k0 * 32 + nt * 16 + lr];
        b.y = sm[O_NFW1 + (k0 + 1) * 32 + nt * 16 + lr];
        acc = wmma4(a, b, acc);
      }
#pragma unroll
      for (int j = 0; j < 8; ++j)
        scr[(j + 8 * half) * 32 + nt * 16 + lr] =
            lrelu(acc[j] + sm[O_NFB1 + nt * 16 + lr]);
    }
    v8f acc2 = {};
#pragma unroll
    for (int kk = 0; kk < 8; ++kk) {  // K = 32
      const int k0 = kk * 4 + half * 2;
      v2f a, b;
      a.x = scr[lr * 32 + k0];
      a.y = scr[lr * 32 + k0 + 1];
      b.x = sm[O_NFW2 + k0 * 16 + lr];
      b.y = sm[O_NFW2 + (k0 + 1) * 16 + lr];
      acc2 = wmma4(a, b, acc2);
    }
    if (lr < 11) {
#pragma unroll
      for (int j = 0; j < 8; ++j) {
        const int node = g * 32 + w * 16 + j + 8 * half;
        out_node[(size_t)node * 11 + lr] = acc2[j] + sm[O_NFB2 + lr];
      }
    }
  }
}

extern "C" void kernel_launch(void* const* d_in, const int* in_sizes, int n_in,
                              void* d_out, int out_size, void* d_ws,
                              size_t ws_size, hipStream_t stream) {
  (void)d_ws;
  (void)ws_size;
  (void)out_size;
  int idx = 0;
  const float* latent = (const float*)d_in[idx++];
  idx++;  // batch (int32) — implicit in group id, unused
  const float* ltw1 = (const float*)d_in[idx++];
  const float* ltw2 = (const float*)d_in[idx++];
  const float *We0, *We1, *be0, *be1, *Wn0, *Wn1, *bn0, *bn1;
  if (n_in >= 24) {
    // per-layer lists passed as separate inputs (3 each; layer 2 unused)
    We0 = (const float*)d_in[idx++];
    We1 = (const float*)d_in[idx++];
    idx++;
    be0 = (const float*)d_in[idx++];
    be1 = (const float*)d_in[idx++];
    idx++;
    Wn0 = (const float*)d_in[idx++];
    Wn1 = (const float*)d_in[idx++];
    idx++;
    bn0 = (const float*)d_in[idx++];
    bn1 = (const float*)d_in[idx++];
    idx++;
  } else {
    // lists concatenated flat per dict entry
    const float* Wea = (const float*)d_in[idx++];
    We0 = Wea;
    We1 = Wea + 37 * 8;  // layer0 is 37x8, layer1 is 40x8
    const float* bea = (const float*)d_in[idx++];
    be0 = bea;
    be1 = bea + 8;
    const float* Wna = (const float*)d_in[idx++];
    Wn0 = Wna;
    Wn1 = Wna + 24 * 16;
    const float* bna = (const float*)d_in[idx++];
    bn0 = bna;
    bn1 = bna + 16;
  }
  const float* etw1 = (const float*)d_in[idx++];
  const float* etb1 = (const float*)d_in[idx++];
  const float* etw2 = (const float*)d_in[idx++];
  const float* etb2 = (const float*)d_in[idx++];
  const float* nfw1 = (const float*)d_in[idx++];
  const float* nfb1 = (const float*)d_in[idx++];
  const float* nfw2 = (const float*)d_in[idx++];
  const float* nfb2 = (const float*)d_in[idx++];

  const int N = in_sizes[0] / 64;  // 65536 nodes
  const int nGroups = N / 32;      // 2048 groups
  float* out_node = (float*)d_out;
  float* out_edge = out_node + (size_t)N * 11;

  const size_t shmem = SMEM_FLOATS * sizeof(float);
  gnn_fused<<<nGroups, 128, shmem, stream>>>(
      latent, ltw1, ltw2, We0, be0, We1, be1, Wn0, bn0, Wn1, bn1, etw1, etb1,
      etw2, etb2, nfw1, nfb1, nfw2, nfb2, out_node, out_edge);
}